// Dynamick_max_pooling1d_9740985827593
// MI455X (gfx1250) — compile-verified
//
#include <hip/hip_runtime.h>
#include <stdint.h>

// x[16, 8192, 512] f32 -> out[16, 64, 512]: per (b,c) keep the 64 largest of 8192,
// in original sequence order (ties resolved like stable ascending argsort take-last).
#define B_      16
#define L_      8192
#define C_      512
#define K_      64
#define CG      4              // channels per block -> 128KB tile, 2 blocks/WGP
#define WAVES   4              // one wave32 per channel
#define TPB     (WAVES * 32)
#define NPL     (L_ / 32)      // elements per lane per channel = 256
#define CAP_E   256            // threshold-bin candidate capacity (N(0,1): ~186 expected)
#define ESLOTS  (CAP_E / 32)
#define RANKCAP 48             // refine candidates below this before O(e^2/32) ranking

// Order-preserving float->u32 key: larger float <=> larger key
__device__ __forceinline__ unsigned orderKey(float f) {
  unsigned u = __float_as_uint(f);
  return (u & 0x80000000u) ? ~u : (u | 0x80000000u);
}

__device__ __forceinline__ unsigned ballot32(bool p) {
#if __has_builtin(__builtin_amdgcn_ballot_w32)
  return __builtin_amdgcn_ballot_w32(p);
#else
  return (unsigned)__ballot(p);
#endif
}

// CDNA5 async global->LDS DMA (ASYNCcnt). GVS mode: SADDR(64b SGPR)+VADDR(32b VGPR).
__device__ __forceinline__ void asyncLoadB128(unsigned ldsAddr, const void* sbase,
                                              unsigned byteOff) {
  asm volatile("global_load_async_to_lds_b128 %0, %1, %2"
               :
               : "v"(ldsAddr), "v"(byteOff), "s"(sbase)
               : "memory");
}
__device__ __forceinline__ void waitAsync0() {
  asm volatile("s_wait_asynccnt 0" ::: "memory");
}
__device__ __forceinline__ void waitDs0() {
  asm volatile("s_wait_dscnt 0" ::: "memory");
}

// Parallel descending scan of a 256-bin histogram: find bin where the
// cumulative-from-top count crosses t. 8 bins/lane + shfl_up wave scan.
__device__ __forceinline__ void scanHist256(const unsigned* hist, int lane, unsigned t,
                                            unsigned& selByte, unsigned& countAbove,
                                            unsigned& binCnt) {
  const int base = 248 - 8 * lane;                       // lane 0 owns top bins
  uint4 u0 = *(const uint4*)(hist + base);
  uint4 u1 = *(const uint4*)(hist + base + 4);
  unsigned cd[8] = {u1.w, u1.z, u1.y, u1.x, u0.w, u0.z, u0.y, u0.x}; // descending bins
  unsigned localSum = 0;
#pragma unroll
  for (int q = 0; q < 8; ++q) localSum += cd[q];
  unsigned incl = localSum;
#pragma unroll
  for (int d = 1; d < 32; d <<= 1) {
    unsigned nb = __shfl_up(incl, d, 32);
    if (lane >= d) incl += nb;
  }
  unsigned excl = incl - localSum;
  bool cross = (excl < t) && (incl >= t);                // exactly one lane
  unsigned mc = ballot32(cross);
  int srcLane = __ffs((int)mc) - 1;
  unsigned sb = 0, ca = 0, bc = 0;
  if (cross) {
    unsigned cum = excl;
#pragma unroll
    for (int q = 0; q < 8; ++q) {
      if (cum + cd[q] >= t) { sb = (unsigned)(255 - 8 * lane - q); ca = cum; bc = cd[q]; break; }
      cum += cd[q];
    }
  }
  selByte    = (unsigned)__shfl((int)sb, srcLane, 32);
  countAbove = (unsigned)__shfl((int)ca, srcLane, 32);
  binCnt     = (unsigned)__shfl((int)bc, srcLane, 32);
}

__global__ __launch_bounds__(TPB)
void kmax_pool_kernel(const float* __restrict__ x, float* __restrict__ out) {
  __shared__ __align__(16) float s_data[L_ * CG];            // 128KB tile [l*CG + ch]
  __shared__ __align__(16) unsigned s_hist[WAVES][256];      // per-wave radix histograms
  __shared__ unsigned short s_ebuf[WAVES][CAP_E];            // threshold-bin candidate indices
  __shared__ unsigned short s_gbuf[WAVES][K_];               // strictly-greater indices (<64)
  __shared__ unsigned short s_sel[WAVES][K_];                // final 64 selected indices
  __shared__ unsigned s_cnt[WAVES][4];                       // [0]=ecnt [1]=gcnt [2]=selcnt

  const int tid  = threadIdx.x;
  const int wave = tid >> 5;
  const int lane = tid & 31;
  const int b    = blockIdx.x / (C_ / CG);
  const int c0   = (blockIdx.x % (C_ / CG)) * CG;
  const float* gbase = x + (size_t)b * L_ * C_ + c0;

  // ---- Stage tile via async DMA: one b128 per row (4 contiguous channels = 16B).
  const unsigned ldsBase = (unsigned)(uintptr_t)&s_data[0];
#pragma unroll 4
  for (int it = 0; it < L_ / TPB; ++it) {                    // uniform trip count: 64
    unsigned n = (unsigned)(tid + it * TPB);
    asyncLoadB128(ldsBase + n * 16u, gbase, n * (C_ * 4));
  }
  waitAsync0();
  __syncthreads();          // only block-wide barrier; waves are independent below

  const int ch = wave;
  if (lane == 0) { s_cnt[wave][0] = 0; s_cnt[wave][1] = 0; s_cnt[wave][2] = 0; }

  // ---- Stage 1: full-data 8-bit radix rounds until threshold bin <= CAP_E.
  unsigned P = 0, M = 0, t = K_, e = L_;
  int shift = 24;
  while (true) {
    for (int j = lane; j < 256; j += 32) s_hist[wave][j] = 0;
    waitDs0();
#pragma unroll 4
    for (int i = 0; i < NPL; ++i) {
      unsigned key = orderKey(s_data[(i * 32 + lane) * CG + ch]);
      if ((key & M) == P) atomicAdd(&s_hist[wave][(key >> shift) & 255u], 1u);
    }
    waitDs0();
    unsigned selByte, above, binCnt;
    scanHist256(s_hist[wave], lane, t, selByte, above, binCnt);
    t -= above;                       // remaining rank inside chosen bin (>=1)
    P |= selByte << shift;
    M |= 0xFFu << shift;
    e = binCnt;
    if (e <= CAP_E || shift == 0) break;
    shift -= 8;
  }

  // ---- Stage 2: single collect pass.  G: (key&M) > P (all selected, <=63).
  //                                     E: (key&M) == P (threshold bin, <=CAP_E).
#pragma unroll 4
  for (int i = 0; i < NPL; ++i) {
    unsigned l = (unsigned)(i * 32 + lane);
    unsigned key = orderKey(s_data[l * CG + ch]);
    unsigned mk = key & M;
    if (mk > P) {
      unsigned slot = atomicAdd(&s_cnt[wave][1], 1u);
      if (slot < K_) s_gbuf[wave][slot] = (unsigned short)l;
    } else if (mk == P) {
      unsigned slot = atomicAdd(&s_cnt[wave][0], 1u);
      if (slot < CAP_E) s_ebuf[wave][slot] = (unsigned short)l;
    }
  }
  waitDs0();
  e = s_cnt[wave][0]; if (e > CAP_E) e = CAP_E;
  unsigned g = s_cnt[wave][1]; if (g > K_) g = K_;

  // ---- Stage 3: cheap candidate-space refinement (tiny N, registers + LDS).
  while (e > RANKCAP && shift > 0) {
    shift -= 8;
    for (int j = lane; j < 256; j += 32) s_hist[wave][j] = 0;
    unsigned myIdx[ESLOTS], myKey[ESLOTS];
#pragma unroll
    for (int s = 0; s < ESLOTS; ++s) {
      unsigned j = (unsigned)lane + 32u * s;
      myIdx[s] = (j < e) ? (unsigned)s_ebuf[wave][j] : 0u;
      myKey[s] = orderKey(s_data[myIdx[s] * CG + ch]);
    }
    waitDs0();
#pragma unroll
    for (int s = 0; s < ESLOTS; ++s) {
      unsigned j = (unsigned)lane + 32u * s;
      if (j < e) atomicAdd(&s_hist[wave][(myKey[s] >> shift) & 255u], 1u);
    }
    waitDs0();
    unsigned selByte, above, binCnt;
    scanHist256(s_hist[wave], lane, t, selByte, above, binCnt);
    t -= above;
    P |= selByte << shift;
    M |= 0xFFu << shift;
    if (lane == 0) s_cnt[wave][0] = 0;
    waitDs0();
#pragma unroll
    for (int s = 0; s < ESLOTS; ++s) {
      unsigned j = (unsigned)lane + 32u * s;
      if (j < e) {
        unsigned mk = myKey[s] & M;
        if (mk > P) {
          unsigned slot = atomicAdd(&s_cnt[wave][1], 1u);
          if (slot < K_) s_gbuf[wave][slot] = (unsigned short)myIdx[s];
        } else if (mk == P) {
          unsigned slot = atomicAdd(&s_cnt[wave][0], 1u);
          s_ebuf[wave][slot] = (unsigned short)myIdx[s];
        }
      }
    }
    waitDs0();
    e = s_cnt[wave][0];
    g = s_cnt[wave][1]; if (g > K_) g = K_;
  }

  // ---- Stage 4: exact top-t of E by composite (key<<13 | index) ranking.
  // Larger index wins ties -> matches stable ascending argsort take-last-K.
  unsigned selIdx[ESLOTS];
  bool selFlag[ESLOTS];
  {
    unsigned long long comp[ESLOTS];
    unsigned cnt[ESLOTS];
    bool valid[ESLOTS];
#pragma unroll
    for (int s = 0; s < ESLOTS; ++s) {
      unsigned j = (unsigned)lane + 32u * s;
      valid[s] = (j < e);
      unsigned idx = valid[s] ? (unsigned)s_ebuf[wave][j] : 0u;
      selIdx[s] = idx;
      unsigned key = orderKey(s_data[idx * CG + ch]);
      comp[s] = (((unsigned long long)key) << 13) | idx;
      cnt[s] = 0;
    }
    if (e > t) {                                   // else everything in E is selected
      for (unsigned j = 0; j < e; ++j) {
        unsigned idxj = s_ebuf[wave][j];           // LDS broadcast
        unsigned keyj = orderKey(s_data[idxj * CG + ch]);
        unsigned long long compj = (((unsigned long long)keyj) << 13) | idxj;
#pragma unroll
        for (int s = 0; s < ESLOTS; ++s) cnt[s] += (compj > comp[s]) ? 1u : 0u;
      }
    }
#pragma unroll
    for (int s = 0; s < ESLOTS; ++s) selFlag[s] = valid[s] && (cnt[s] < t);
  }

  // ---- Stage 5: gather 64 selected indices; position = rank among them.
  for (unsigned i = (unsigned)lane; i < g; i += 32) s_sel[wave][i] = s_gbuf[wave][i];
  if (lane == 0) s_cnt[wave][2] = g;
  waitDs0();
#pragma unroll
  for (int s = 0; s < ESLOTS; ++s) {
    if (selFlag[s]) {
      unsigned slot = atomicAdd(&s_cnt[wave][2], 1u);
      if (slot < K_) s_sel[wave][slot] = (unsigned short)selIdx[s];
    }
  }
  waitDs0();
  unsigned total = s_cnt[wave][2]; if (total > K_) total = K_;

  unsigned a0 = ((unsigned)lane < total)       ? (unsigned)s_sel[wave][lane]      : 0xFFFFu;
  unsigned a1 = ((unsigned)(lane + 32) < total) ? (unsigned)s_sel[wave][lane + 32] : 0xFFFFu;
  unsigned p0 = 0, p1 = 0;
  for (unsigned j = 0; j < total; ++j) {
    unsigned ij = s_sel[wave][j];                  // LDS broadcast
    p0 += (ij < a0) ? 1u : 0u;
    p1 += (ij < a1) ? 1u : 0u;
  }
  float* obase = out + (size_t)b * K_ * C_ + (unsigned)(c0 + ch);
  if ((unsigned)lane < total)        obase[(size_t)p0 * C_] = s_data[a0 * CG + ch];
  if ((unsigned)(lane + 32) < total) obase[(size_t)p1 * C_] = s_data[a1 * CG + ch];
}

extern "C" void kernel_launch(void* const* d_in, const int* in_sizes, int n_in,
                              void* d_out, int out_size, void* d_ws, size_t ws_size,
                              hipStream_t stream) {
  (void)in_sizes; (void)n_in; (void)d_ws; (void)ws_size; (void)out_size;
  const float* x = (const float*)d_in[0];
  float* out     = (float*)d_out;
  const int nblocks = B_ * (C_ / CG);   // 2048 blocks: (batch, 4-channel group)
  kmax_pool_kernel<<<dim3(nblocks), dim3(TPB), 0, stream>>>(x, out);
}